// Router_73031623901859
// MI455X (gfx1250) — compile-verified
//
#include <hip/hip_runtime.h>
#include <math.h>

// ---- problem constants (match reference: B=4, S=4096, H=4096, E=64, k=2) ----
#define HDIM   4096
#define NEXP   64
#define TOPK   2
#define KC     128            // K-chunk staged in LDS
#define NCHUNK (HDIM / KC)    // 32
#define PAD    4              // LDS row pad (floats) -> conflict-free ds_load_b64
#define LROW   (KC + PAD)     // 132 floats per LDS row
#define WPB    4              // waves per block
#define TPB    (WPB * 32)     // 128 threads per block

typedef __attribute__((ext_vector_type(2))) float v2f;
typedef __attribute__((ext_vector_type(8))) float v8f;

// One async-to-LDS b128 per macro: moves 16B per lane, global -> LDS, ASYNCcnt.
// Same IOFFSET is applied to both the LDS and global addresses (ISA 08 §4.4).
#define ASYNC_LD16(ldsa, gaddr, OFS)                                          \
    asm volatile("global_load_async_to_lds_b128 %0, %1, off offset:" OFS      \
                 :: "v"(ldsa), "v"(gaddr) : "memory")

// Copy 256B per lane (16 x b128) from global to LDS asynchronously.
__device__ __forceinline__ void async_copy_256B(unsigned ldsa, unsigned long long g) {
    ASYNC_LD16(ldsa, g, "0");   ASYNC_LD16(ldsa, g, "16");
    ASYNC_LD16(ldsa, g, "32");  ASYNC_LD16(ldsa, g, "48");
    ASYNC_LD16(ldsa, g, "64");  ASYNC_LD16(ldsa, g, "80");
    ASYNC_LD16(ldsa, g, "96");  ASYNC_LD16(ldsa, g, "112");
    ASYNC_LD16(ldsa, g, "128"); ASYNC_LD16(ldsa, g, "144");
    ASYNC_LD16(ldsa, g, "160"); ASYNC_LD16(ldsa, g, "176");
    ASYNC_LD16(ldsa, g, "192"); ASYNC_LD16(ldsa, g, "208");
    ASYNC_LD16(ldsa, g, "224"); ASYNC_LD16(ldsa, g, "240");
}

__device__ __forceinline__ void wait_async0(void) {
    asm volatile("s_wait_asynccnt 0x0" ::: "memory");
}

__device__ __forceinline__ unsigned lds_offset_u32(const void* p) {
    // Low 32 bits of a shared-memory flat address are the LDS byte offset.
    return (unsigned)(__SIZE_TYPE__)(__attribute__((address_space(3))) const char*)p;
}

// One block: 4 waves x 16 tokens = 64 tokens, all 64 experts.
// W K-chunks are staged in LDS (double buffered, async DMA) and shared by the waves.
__global__ __launch_bounds__(TPB)
void router_gemm_kernel(const float* __restrict__ X, const float* __restrict__ W,
                        const float* __restrict__ bias, float* __restrict__ logits)
{
    __shared__ float bsm[2][NEXP * LROW];   // 2 x 33.8 KB

    const int lane = (int)(threadIdx.x & 31);
    const int wid  = (int)(threadIdx.x >> 5);               // wave in block
    const int m    = lane & 15;                             // A row / expert-in-group
    const int hi   = lane >> 4;                             // K-half per ISA A/B layout
    const int row0 = ((int)blockIdx.x * WPB + wid) * 16;    // first token of this wave

    // ---- staging assignment: thread -> (W row, 64-float half) => 16 x async b128 ----
    const int srow  = (int)(threadIdx.x >> 1);              // 0..63
    const int shalf = (int)(threadIdx.x & 1) * 64;          // 0 or 64
    const float*   sgl   = W + (size_t)srow * HDIM + shalf; // + c*KC per chunk
    const unsigned sld0  = lds_offset_u32(&bsm[0][srow * LROW + shalf]);
    const unsigned sld1  = lds_offset_u32(&bsm[1][srow * LROW + shalf]);

    // ---- per-wave operand bases ----
    const float* aP   = X + (size_t)(row0 + m) * HDIM + 2 * hi;   // global A
    const int    bofs = m * LROW + 2 * hi;                        // LDS B base

    v8f acc0 = {}, acc1 = {}, acc2 = {}, acc3 = {};

    // preload chunk 0 via async DMA
    async_copy_256B(sld0, (unsigned long long)(const void*)sgl);
    wait_async0();
    __syncthreads();

    for (int c = 0; c < NCHUNK; ++c) {
        const int buf = c & 1;

        // kick off async DMA for the next chunk into the other buffer
        if (c + 1 < NCHUNK) {
            async_copy_256B(buf ? sld0 : sld1,
                            (unsigned long long)(const void*)(sgl + (c + 1) * KC));
        }

        // compute on current chunk: A from global (streamed once), B from LDS
        const float* a = aP + c * KC;
        const float* b = &bsm[buf][bofs];
#pragma unroll 8
        for (int kk = 0; kk < KC; kk += 4) {
            v2f av  = *(const v2f*)(a + kk);
            v2f bv0 = *(const v2f*)(b + kk);
            v2f bv1 = *(const v2f*)(b + 16 * LROW + kk);
            v2f bv2 = *(const v2f*)(b + 32 * LROW + kk);
            v2f bv3 = *(const v2f*)(b + 48 * LROW + kk);
            acc0 = __builtin_amdgcn_wmma_f32_16x16x4_f32(false, av, false, bv0, (short)0, acc0, false, false);
            acc1 = __builtin_amdgcn_wmma_f32_16x16x4_f32(false, av, false, bv1, (short)0, acc1, false, false);
            acc2 = __builtin_amdgcn_wmma_f32_16x16x4_f32(false, av, false, bv2, (short)0, acc2, false, false);
            acc3 = __builtin_amdgcn_wmma_f32_16x16x4_f32(false, av, false, bv3, (short)0, acc3, false, false);
        }

        // my async transfers done, then block-wide visibility
        wait_async0();
        __syncthreads();
    }

    // ---- bias add + store ----
    const float bn0 = bias[0 * 16 + m];
    const float bn1 = bias[1 * 16 + m];
    const float bn2 = bias[2 * 16 + m];
    const float bn3 = bias[3 * 16 + m];

    // C/D layout: VGPR r holds M = r (lanes 0-15) or M = r+8 (lanes 16-31), N = lane&15.
#pragma unroll
    for (int r = 0; r < 8; ++r) {
        const int token = row0 + r + 8 * hi;
        float* out = logits + (size_t)token * NEXP + m;
        out[0]  = acc0[r] + bn0;
        out[16] = acc1[r] + bn1;
        out[32] = acc2[r] + bn2;
        out[48] = acc3[r] + bn3;
    }
}

// Per-token top-2 + softmax over the two selected logits (4 MB read: negligible).
__global__ __launch_bounds__(256)
void router_topk_kernel(const float* __restrict__ logits,
                        float* __restrict__ idx_out,
                        float* __restrict__ w_out, int T)
{
    const int t = (int)(blockIdx.x * blockDim.x + threadIdx.x);
    if (t >= T) return;
    const float* row = logits + (size_t)t * NEXP;

    float v0 = -INFINITY; int i0 = 0;
#pragma unroll 8
    for (int i = 0; i < NEXP; ++i) {
        float v = row[i];
        if (v > v0) { v0 = v; i0 = i; }
    }
    float v1 = -INFINITY; int i1 = 0;
#pragma unroll 8
    for (int i = 0; i < NEXP; ++i) {
        if (i == i0) continue;
        float v = row[i];
        if (v > v1) { v1 = v; i1 = i; }
    }
    // softmax([v0, v1]) with v0 >= v1
    const float e  = __expf(v1 - v0);
    const float w0 = 1.0f / (1.0f + e);

    idx_out[2 * t]     = (float)i0;
    idx_out[2 * t + 1] = (float)i1;
    w_out[2 * t]       = w0;
    w_out[2 * t + 1]   = 1.0f - w0;
}

extern "C" void kernel_launch(void* const* d_in, const int* in_sizes, int n_in,
                              void* d_out, int out_size, void* d_ws, size_t ws_size,
                              hipStream_t stream) {
    const float* X    = (const float*)d_in[0];   // [B,S,H] f32
    const float* W    = (const float*)d_in[1];   // [E,H]   f32
    const float* bias = (const float*)d_in[2];   // [E]     f32
    // d_in[3] is k (==2), fixed at compile time.

    const int E = in_sizes[2];                   // 64
    const int H = in_sizes[1] / E;               // 4096
    const int T = in_sizes[0] / H;               // 16384 tokens

    float* logits = (float*)d_out;               // [T, E]
    float* idx    = logits + (size_t)T * E;      // [T, 2] (indices stored as f32)
    float* wts    = idx + (size_t)T * TOPK;      // [T, 2]

    const int blocks = T / (WPB * 16);           // 256 blocks x 4 waves x 16 tokens
    router_gemm_kernel<<<blocks, TPB, 0, stream>>>(X, W, bias, logits);

    router_topk_kernel<<<(T + 255) / 256, 256, 0, stream>>>(logits, idx, wts, T);
}